// PCENLayer_77824807403757
// MI455X (gfx1250) — compile-verified
//
#include <hip/hip_runtime.h>

typedef __attribute__((ext_vector_type(2))) float v2f;
typedef __attribute__((ext_vector_type(8))) float v8f;

#define ALPHA_F    0.98f
#define ONEMA_F    0.02f
#define L2ALPHA   -0.02914634565f   // log2(0.98)
#define T          4000
#define RPW        16               // rows per workgroup
#define NWAVES     10
#define SEG        400              // timesteps per wave segment
#define NBLOCKS    25               // 16-step WMMA blocks per segment
#define LDS_STRIDE 4001             // padded row stride (dwords) to spread banks
#define EPS_F      1e-6f
#define SQRT_DELTA 1.41421356237f   // sqrt(2.0)

__device__ __forceinline__ float apow(float d) {  // alpha^d, d >= 0
    return __builtin_amdgcn_exp2f(d * L2ALPHA);
}

__global__ __launch_bounds__(320) void pcen_wmma_kernel(const float* __restrict__ x,
                                                        float* __restrict__ out) {
    extern __shared__ float smem[];
    float* s0 = smem;                              // RPW * LDS_STRIDE : zero-carry smooth
    float* pb = smem + RPW * LDS_STRIDE;           // NWAVES*RPW : segment-end partials
    float* cw = pb + NWAVES * RPW;                 // NWAVES*RPW : exclusive carries

    const int tid  = threadIdx.x;
    const int lane = tid & 31;
    const int wave = tid >> 5;       // 0..9
    const int mm   = lane & 15;      // row (B/C/D column), or M (A row)
    const int half = lane >> 4;      // lane half selects K pairs / M+8

    const size_t base = (size_t)blockIdx.x * RPW * T;  // 16 contiguous rows
    const float* xg = x + base;
    float*       og = out + base;

    // ---- constant coefficient matrix A[m][k] = (k<=m) ? 0.02*0.98^(m-k) : 0
    // f32 16x4 A layout: lane m -> K = 4kk+{0,1}; lane m+16 -> K = 4kk+{2,3}
    v2f A[4];
#pragma unroll
    for (int kk = 0; kk < 4; ++kk) {
        int k0 = kk * 4 + half * 2;
        A[kk].x = (k0     <= mm) ? ONEMA_F * apow((float)(mm - k0))     : 0.0f;
        A[kk].y = (k0 + 1 <= mm) ? ONEMA_F * apow((float)(mm - k0 - 1)) : 0.0f;
    }
    // global block 0: s_0 = x_0 exactly -> column k=0 coefficient is alpha^m
    v2f A0c0 = A[0];
    if (half == 0) A0c0.x = apow((float)mm);

    // carry-fold multipliers alpha^(M+1) for each accumulator VGPR (M = j + 8*half)
    float pav[8];
#pragma unroll
    for (int j = 0; j < 8; ++j) pav[j] = apow((float)(j + 1 + 8 * half));

    const float* xrow = xg + mm * T;

    // ---- phase 1: chained 16x16 WMMA block-scan over this wave's segment
    float carry = 0.0f;
    const int tseg = wave * SEG;
    for (int blk = 0; blk < NBLOCKS; ++blk) {
        const int t0 = tseg + blk * 16;
        // C init = alpha^(M+1) * carry  (carry contribution, affine linearity)
        v8f d;
#pragma unroll
        for (int j = 0; j < 8; ++j) d[j] = pav[j] * carry;

        const bool first = (wave == 0) && (blk == 0);   // wave-uniform, EXEC stays full
#pragma unroll
        for (int kk = 0; kk < 4; ++kk) {
            v2f a = A[kk];
            if (kk == 0 && first) a = A0c0;
            // B 4x16 f32 layout: lane n -> K={4kk,4kk+1}; lane n+16 -> K={4kk+2,4kk+3}
            const float* p = xrow + t0 + kk * 4 + half * 2;
            v2f b; b.x = p[0]; b.y = p[1];
            d = __builtin_amdgcn_wmma_f32_16x16x4_f32(false, a, false, b,
                                                      (short)0, d, false, false);
        }
        // D layout: VGPR j -> (M=j, N=lane) / (M=j+8, N=lane-16); write smooth to LDS
#pragma unroll
        for (int j = 0; j < 8; ++j)
            s0[mm * LDS_STRIDE + t0 + j + 8 * half] = d[j];
        // next intra-wave carry: s(M=15) lives in VGPR7 lanes 16..31 -> broadcast
        carry = __shfl(d[7], mm + 16, 32);
    }
    if (lane < 16) pb[wave * 16 + mm] = carry;  // segment partial b_w (zero carry-in)
    __syncthreads();

    // ---- phase 2: tiny exclusive affine scan over the 10 wave partials per row
    if (tid < 16) {
        const float a_seg = apow((float)SEG);   // 0.98^400 ~ 3.09e-4
        float c = 0.0f;
        cw[tid] = 0.0f;
        for (int w = 1; w < NWAVES; ++w) {
            c = __builtin_fmaf(a_seg, c, pb[(w - 1) * 16 + tid]);
            cw[w * 16 + tid] = c;
        }
    }
    __syncthreads();

    // ---- phase 3: coalesced fix-up + PCEN epilogue (x re-read hits L2)
    for (int e = tid; e < RPW * T; e += 320) {
        int row = e / T;
        int t   = e - row * T;
        int w   = t / SEG;
        int off = t - w * SEG;
        float s  = __builtin_fmaf(apow((float)(off + 1)), cw[w * 16 + row],
                                  s0[row * LDS_STRIDE + t]);
        float sm = s + EPS_F;
        // (sm)^-0.98 via v_log_f32 (log2) + v_exp_f32 (exp2); avoids a divide
        float pinv = __builtin_amdgcn_exp2f(-ALPHA_F * __builtin_amdgcn_logf(sm));
        float r = __builtin_amdgcn_sqrtf(__builtin_fmaf(xg[e], pinv, 2.0f)) - SQRT_DELTA;
        og[e] = r;
    }
}

extern "C" void kernel_launch(void* const* d_in, const int* in_sizes, int n_in,
                              void* d_out, int out_size, void* d_ws, size_t ws_size,
                              hipStream_t stream) {
    (void)n_in; (void)d_ws; (void)ws_size; (void)out_size;
    const float* x = (const float*)d_in[0];
    float* out = (float*)d_out;

    const int total = in_sizes[0];          // 32*128*4000
    const int rows  = total / T;            // 4096
    const int grid  = rows / RPW;           // 256 workgroups

    const size_t lds_bytes =
        (size_t)(RPW * LDS_STRIDE + 2 * NWAVES * RPW) * sizeof(float);  // ~257 KB (<320 KB/WGP)

    pcen_wmma_kernel<<<grid, 320, lds_bytes, stream>>>(x, out);
}